// Self_Attention_67388036874861
// MI455X (gfx1250) — compile-verified
//
#include <hip/hip_runtime.h>

typedef float v2f __attribute__((ext_vector_type(2)));
typedef float v8f __attribute__((ext_vector_type(8)));

#define B_   4
#define S_   2048
#define HID_ 512
#define NH_  8
#define DH_  64
#define M_TOT (B_ * S_)   // 8192

// CDNA5 f32 WMMA: D(16x16,f32) = A(16x4,f32) * B(4x16,f32) + C
// A frag: lane L: row = L%16; v0/v1 hold K = (L/16)*2 + {0,1}
// B frag: lane L: col = L%16; v0/v1 hold K = (L/16)*2 + {0,1}
// C/D   : lane L: col = L%16; reg r holds row = r + 8*(L/16)
__device__ __forceinline__ v8f wmma4(v2f a, v2f b, v8f c) {
  return __builtin_amdgcn_wmma_f32_16x16x4_f32(
      /*neg_a=*/false, a, /*neg_b=*/false, b,
      /*c_mod=*/(short)0, c, /*reuse_a=*/false, /*reuse_b=*/false);
}

__device__ __forceinline__ v8f vzero8() {
  v8f z;
#pragma unroll
  for (int i = 0; i < 8; ++i) z[i] = 0.f;
  return z;
}

// ---- CDNA5 async global->LDS (ASYNCcnt) staging helpers -------------------
__device__ __forceinline__ unsigned lds_off(const void* p) {
  // generic -> LDS addrspace cast; AS(3) pointer value is the LDS byte offset
  return (unsigned)(unsigned long long)(
      (__attribute__((address_space(3))) const void*)p);
}
__device__ __forceinline__ void async_copy16(unsigned lds_byte,
                                             const float* g) {
  asm volatile("global_load_async_to_lds_b128 %0, %1, off"
               :
               : "v"(lds_byte), "v"(g)
               : "memory");
}
// ---------------------------------------------------------------------------

// out[M,512] = A[M,512] @ W[512,512] + bias (+ resid).
// Block tile 256x64, 8 waves, wave tile 32x64 (2 A-frags share each B-frag).
// Double-buffered async pipeline: stage chunk k+1 while computing chunk k.
__global__ __launch_bounds__(256) void gemm_bias_kernel(
    const float* __restrict__ A, const float* __restrict__ W,
    const float* __restrict__ bias, const float* __restrict__ resid,
    float* __restrict__ out) {
  // stride 36 floats: rows 16B-aligned (async B128 ok) and 36*m % 64 injective
  // for m=0..15 -> conflict-free 16-lane column reads of A fragments.
  __shared__ float as_[2][256][36];
  __shared__ float ws_[2][32][64];

  const int m0 = blockIdx.x * 256;
  const int n0 = blockIdx.y * 64;
  const int tid = threadIdx.x;
  const int wave = tid >> 5, lane = tid & 31;
  const int lh = lane >> 4, ln = lane & 15;

  v8f acc[2][4];
#pragma unroll
  for (int i = 0; i < 2; ++i)
#pragma unroll
    for (int j = 0; j < 4; ++j) acc[i][j] = vzero8();

  // 10 async ops per thread per chunk (8 for A tile, 2 for W tile)
  auto stage = [&](int buf, int k0) {
#pragma unroll
    for (int r = 0; r < 8; ++r) {
      int idx = tid + r * 256;
      int row = idx >> 3, col = (idx & 7) << 2;
      async_copy16(lds_off(&as_[buf][row][col]),
                   A + (size_t)(m0 + row) * 512 + k0 + col);
    }
#pragma unroll
    for (int r = 0; r < 2; ++r) {
      int idx = tid + r * 256;
      int row = idx >> 4, col = (idx & 15) << 2;
      async_copy16(lds_off(&ws_[buf][row][col]),
                   W + (size_t)(k0 + row) * 512 + n0 + col);
    }
  };

  stage(0, 0);
  for (int kc = 0; kc < 16; ++kc) {
    if (kc + 1 < 16) {
      // buf (kc+1)&1 was last read in iteration kc-1; the barrier at the end
      // of that iteration makes this overwrite safe.
      stage((kc + 1) & 1, (kc + 1) * 32);
      // async ops complete in order: <=10 outstanding means chunk kc landed
      asm volatile("s_wait_asynccnt 0xa" ::: "memory");
    } else {
      asm volatile("s_wait_asynccnt 0x0" ::: "memory");
    }
    __syncthreads();

    const float(*as)[36] = as_[kc & 1];
    const float(*ws)[64] = ws_[kc & 1];
    const int mr0 = wave * 32 + ln;
#pragma unroll
    for (int kk = 0; kk < 32; kk += 4) {
      v2f a0, a1;
      a0.x = as[mr0][kk + lh * 2];
      a0.y = as[mr0][kk + lh * 2 + 1];
      a1.x = as[mr0 + 16][kk + lh * 2];
      a1.y = as[mr0 + 16][kk + lh * 2 + 1];
#pragma unroll
      for (int nt = 0; nt < 4; ++nt) {
        v2f b;
        b.x = ws[kk + lh * 2][nt * 16 + ln];
        b.y = ws[kk + lh * 2 + 1][nt * 16 + ln];
        acc[0][nt] = wmma4(a0, b, acc[0][nt]);
        acc[1][nt] = wmma4(a1, b, acc[1][nt]);
      }
    }
    __syncthreads();
  }

  const bool hasres = (resid != nullptr);
#pragma unroll
  for (int half = 0; half < 2; ++half) {
    const int mbase = m0 + wave * 32 + half * 16 + lh * 8;
#pragma unroll
    for (int nt = 0; nt < 4; ++nt) {
      const int col = n0 + nt * 16 + ln;
      const float bb = bias[col];
#pragma unroll
      for (int r = 0; r < 8; ++r) {
        size_t o = (size_t)(mbase + r) * 512 + col;
        float vv = acc[half][nt][r] + bb;
        if (hasres) vv += resid[o];
        out[o] = vv;
      }
    }
  }
}

// Flash attention per (i-tile=128, head, batch). NOTE: reference uses v as
// keys. RoPE is fused into the q-fragment load: pair (d, d+32) lives in the
// same lane as fragments qa[t] / qa[t+8].
__global__ __launch_bounds__(256) void attn_kernel(
    const float* __restrict__ q, const float* __restrict__ v,
    float* __restrict__ hout) {
  // stride 68: rows 16B-aligned for async B128; 68*m % 64 = 4m injective over
  // m=0..15 -> conflict-free column reads for score B-fragments.
  __shared__ float vs[2][64][68];
  __shared__ float ps[8][16][65];  // per-wave P tile (C-layout -> A-layout)

  const int it = blockIdx.x;  // 16
  const int h = blockIdx.y;   // 8
  const int b = blockIdx.z;   // 4
  const int tid = threadIdx.x;
  const int wave = tid >> 5, lane = tid & 31;
  const int lh = lane >> 4, ln = lane & 15;
  const float scale = 0.125f;  // 64^-0.5

  const size_t rowbase = (size_t)b * S_ + it * 128;
  const int srow = it * 128 + wave * 16 + ln;  // sequence position of A-row

  // q A-fragments in registers (16 k-steps over DH=64) + fused RoPE
  v2f qa[16];
  {
    const float* qrow = q + (rowbase + wave * 16 + ln) * 512 + h * 64 + lh * 2;
#pragma unroll
    for (int t = 0; t < 16; ++t) qa[t] = *(const v2f*)(qrow + t * 4);
#pragma unroll
    for (int t = 0; t < 8; ++t) {
#pragma unroll
      for (int c = 0; c < 2; ++c) {
        int d = t * 4 + lh * 2 + c;  // 0..31
        float theta = __powf(10000.f, -(float)d * (1.0f / 32.0f));
        float ang = (float)srow * theta;
        float cs = __cosf(ang), sn = __sinf(ang);
        float x1 = qa[t][c], x2 = qa[t + 8][c];
        qa[t][c] = x1 * cs - x2 * sn;
        qa[t + 8][c] = x2 * cs + x1 * sn;
      }
    }
  }

  v8f hacc[4];
#pragma unroll
  for (int i = 0; i < 4; ++i) hacc[i] = vzero8();
  float mrun[8], lrun[8];
#pragma unroll
  for (int r = 0; r < 8; ++r) { mrun[r] = -3.0e38f; lrun[r] = 0.f; }

  // 4 async ops per thread per v tile
  auto stagev = [&](int buf, int jtile) {
#pragma unroll
    for (int r = 0; r < 4; ++r) {
      int idx = tid + r * 256;
      int row = idx >> 4, col = (idx & 15) << 2;
      async_copy16(lds_off(&vs[buf][row][col]),
                   v + ((size_t)b * S_ + jtile * 64 + row) * 512 + h * 64 + col);
    }
  };

  stagev(0, 0);
  for (int jt = 0; jt < 32; ++jt) {
    if (jt + 1 < 32) {
      stagev((jt + 1) & 1, jt + 1);
      asm volatile("s_wait_asynccnt 0x4" ::: "memory");
    } else {
      asm volatile("s_wait_asynccnt 0x0" ::: "memory");
    }
    __syncthreads();
    const float(*vt)[68] = vs[jt & 1];

    // scores s = q . v^T  (4 column tiles of 16)
    v8f s[4];
#pragma unroll
    for (int i = 0; i < 4; ++i) s[i] = vzero8();
#pragma unroll
    for (int t = 0; t < 16; ++t) {
#pragma unroll
      for (int nt = 0; nt < 4; ++nt) {
        v2f bf;
        bf.x = vt[nt * 16 + ln][t * 4 + lh * 2];
        bf.y = vt[nt * 16 + ln][t * 4 + lh * 2 + 1];
        s[nt] = wmma4(qa[t], bf, s[nt]);
      }
    }
#pragma unroll
    for (int nt = 0; nt < 4; ++nt)
#pragma unroll
      for (int r = 0; r < 8; ++r) s[nt][r] *= scale;

    // online softmax per row (row = r + 8*lh; cols across 16 lanes x 4 tiles)
#pragma unroll
    for (int r = 0; r < 8; ++r) {
      float mx = fmaxf(fmaxf(s[0][r], s[1][r]), fmaxf(s[2][r], s[3][r]));
#pragma unroll
      for (int off = 1; off < 16; off <<= 1)
        mx = fmaxf(mx, __shfl_xor(mx, off, 32));
      float mnew = fmaxf(mrun[r], mx);
      float corr = __expf(mrun[r] - mnew);
      mrun[r] = mnew;
      float rs = 0.f;
#pragma unroll
      for (int nt = 0; nt < 4; ++nt) {
        float p = __expf(s[nt][r] - mnew);
        s[nt][r] = p;
        rs += p;
      }
#pragma unroll
      for (int off = 1; off < 16; off <<= 1) rs += __shfl_xor(rs, off, 32);
      lrun[r] = lrun[r] * corr + rs;
#pragma unroll
      for (int dt = 0; dt < 4; ++dt) hacc[dt][r] *= corr;
    }

    // C-layout -> A-layout via per-wave LDS (same-wave DS ops are in order)
#pragma unroll
    for (int nt = 0; nt < 4; ++nt)
#pragma unroll
      for (int r = 0; r < 8; ++r)
        ps[wave][r + 8 * lh][nt * 16 + ln] = s[nt][r];

    // h += P @ v_tile  (K = 64)
#pragma unroll
    for (int t = 0; t < 16; ++t) {
      v2f af;
      af.x = ps[wave][ln][t * 4 + lh * 2];
      af.y = ps[wave][ln][t * 4 + lh * 2 + 1];
#pragma unroll
      for (int dt = 0; dt < 4; ++dt) {
        v2f bf;
        bf.x = vt[t * 4 + lh * 2][dt * 16 + ln];
        bf.y = vt[t * 4 + lh * 2 + 1][dt * 16 + ln];
        hacc[dt] = wmma4(af, bf, hacc[dt]);
      }
    }
    __syncthreads();  // vs[jt&1] reads done before it is restaged at jt+1
  }

  // normalize and write h[b,i,h*64+d]
#pragma unroll
  for (int dt = 0; dt < 4; ++dt)
#pragma unroll
    for (int r = 0; r < 8; ++r) {
      size_t row = rowbase + wave * 16 + 8 * lh + r;
      hout[row * 512 + h * 64 + dt * 16 + ln] = hacc[dt][r] / lrun[r];
    }
}

extern "C" void kernel_launch(void* const* d_in, const int* in_sizes, int n_in,
                              void* d_out, int out_size, void* d_ws,
                              size_t ws_size, hipStream_t stream) {
  const float* x  = (const float*)d_in[0];
  const float* Wq = (const float*)d_in[1];
  const float* bq = (const float*)d_in[2];
  // d_in[3]=Wk, d_in[4]=bk: unused (reference never uses k)
  const float* Wv = (const float*)d_in[5];
  const float* bv = (const float*)d_in[6];
  const float* Wo = (const float*)d_in[7];
  const float* bo = (const float*)d_in[8];
  float* out = (float*)d_out;

  float* qb = (float*)d_ws;                 // [8192,512] raw q (rope fused later)
  float* vb = qb + (size_t)M_TOT * 512;     // [8192,512]
  float* hb = vb + (size_t)M_TOT * 512;     // [8192,512]

  dim3 blk(256);
  gemm_bias_kernel<<<dim3(32, 8), blk, 0, stream>>>(x, Wq, bq, nullptr, qb);
  gemm_bias_kernel<<<dim3(32, 8), blk, 0, stream>>>(x, Wv, bv, nullptr, vb);
  attn_kernel<<<dim3(16, 8, 4), blk, 0, stream>>>(qb, vb, hb);
  gemm_bias_kernel<<<dim3(32, 8), blk, 0, stream>>>(hb, Wo, bo, x, out);
}